// AntisymmetricRNN_12292196401859
// MI455X (gfx1250) — compile-verified
//
#include <hip/hip_runtime.h>
#include <hip/hip_bf16.h>

#define UNITS   256
#define FT_DIM  128
#define BATCH   32
#define TSTEPS  2048
#define EPSILON 0.01f
#define GAMMA   0.01f
#define LDSTRIDE 260   // 256 + 4 pad: de-aligns LDS banks for per-lane b128 column reads

typedef __attribute__((ext_vector_type(2))) float v2f;
typedef __attribute__((ext_vector_type(8))) float v8f;

// ---------------------------------------------------------------------------
// Phase 1: h = x @ V + bias, written into d_out (consumed in-place by scan).
// Full fp32 precision via V_WMMA_F32_16X16X4_F32; 32 chained WMMAs per tile.
// ---------------------------------------------------------------------------
__global__ void __launch_bounds__(256)
asym_rnn_proj_wmma(const float* __restrict__ X,     // [65536, 128]
                   const float* __restrict__ V,     // [128, 256]
                   const float* __restrict__ bias,  // [256]
                   float* __restrict__ H)           // [65536, 256]
{
    const int lane   = threadIdx.x & 31;
    const int waveId = (blockIdx.x << 3) | (threadIdx.x >> 5);  // 0..65535
    const int tileRow = waveId >> 4;   // 0..4095
    const int tileCol = waveId & 15;   // 0..15

    const int lhalf = lane >> 4;       // 0|1
    const int lmod  = lane & 15;

    const int rowBase = tileRow << 4;
    const int aRow    = rowBase + lmod;        // A row this lane supplies
    const int bCol    = (tileCol << 4) + lmod; // B/C column this lane owns

    const float* __restrict__ Xr = X + (size_t)aRow * FT_DIM;

    v8f c = {};
#pragma unroll
    for (int k = 0; k < FT_DIM; k += 4) {
        const int ks = k + (lhalf << 1);
        v2f a, b;
        a.x = Xr[ks];
        a.y = Xr[ks + 1];
        b.x = V[(size_t)ks * UNITS + bCol];
        b.y = V[(size_t)(ks + 1) * UNITS + bCol];
        c = __builtin_amdgcn_wmma_f32_16x16x4_f32(
                /*neg_a=*/false, a, /*neg_b=*/false, b,
                /*c_mod=*/(short)0, c, /*reuse_a=*/false, /*reuse_b=*/false);
    }

    const float bv = bias[bCol];
#pragma unroll
    for (int r = 0; r < 8; ++r) {
        const int m = r + (lhalf << 3);
        H[(size_t)(rowBase + m) * UNITS + bCol] = c[r] + bv;
    }
}

// ---------------------------------------------------------------------------
// Phase 2: sequential scan, one workgroup per batch element.
// M = W - W^T - gamma*I lives transposed+padded in LDS (~262 KB -- CDNA5's
// 320 KB WGP LDS only). Double-buffered state -> ONE barrier per step.
// h_{t+1} register-prefetched across the barrier. In-place h -> state.
// ---------------------------------------------------------------------------
__global__ void __launch_bounds__(256)
asym_rnn_scan(const float* __restrict__ W,   // [256,256]
              const float* __restrict__ x0,  // [256]
              float* __restrict__ HS)        // [32, 2048, 256] (h in, states out)
{
    extern __shared__ float smem[];
    float* Mt  = smem;                          // [256][LDSTRIDE] Mt[u][k] = M[k][u]
    float* stA = smem + UNITS * LDSTRIDE;       // [256] state ping
    float* stB = stA + UNITS;                   // [256] state pong

    const int tid = threadIdx.x;
    const int b   = blockIdx.x;

    // Build Mt[u][k] = W[k][u] - W[u][k] - (k==u)*gamma (coalesced W[k][:] reads).
    for (int i = tid; i < UNITS * UNITS; i += 256) {
        const int u = i & (UNITS - 1);
        const int k = i >> 8;
        float m = W[(size_t)k * UNITS + u] - W[(size_t)u * UNITS + k];
        if (k == u) m -= GAMMA;
        Mt[u * LDSTRIDE + k] = m;
    }
    stA[tid] = x0[tid];
    __syncthreads();

    const float* __restrict__ Mrow = Mt + tid * LDSTRIDE;  // 16B aligned
    float* __restrict__ hp = HS + ((size_t)b * TSTEPS) * UNITS + tid;

    float hCur = hp[0];

    auto step = [&](int t, const float* __restrict__ sc, float* __restrict__ sn) {
        // prefetch next h before the long dot product (guarded on last step)
        const float hNext = (t + 1 < TSTEPS) ? hp[(size_t)(t + 1) * UNITS] : 0.0f;

        // dot(state, M[:,tid]) with 4 accumulators; b128 LDS loads on both sides
        float a0 = 0.f, a1 = 0.f, a2 = 0.f, a3 = 0.f;
#pragma unroll
        for (int k = 0; k < UNITS; k += 4) {
            const float4 sv = *(const float4*)(sc + k);      // wave-broadcast read
            const float4 mv = *(const float4*)(Mrow + k);    // per-lane column
            a0 = fmaf(sv.x, mv.x, a0);
            a1 = fmaf(sv.y, mv.y, a1);
            a2 = fmaf(sv.z, mv.z, a2);
            a3 = fmaf(sv.w, mv.w, a3);
        }
        float z = hCur + ((a0 + a1) + (a2 + a3));

        // branch-free tanh: clamp so exp never overflows (tanh(15)==1 in f32)
        z = fminf(fmaxf(z, -15.0f), 15.0f);
        const float e  = __expf(2.0f * z);
        const float th = (e - 1.0f) / (e + 1.0f);

        const float sNew = sc[tid] + EPSILON * th;

        sn[tid] = sNew;                  // only read after the barrier below
        hp[(size_t)t * UNITS] = sNew;    // overwrite h_t with state_t
        __syncthreads();                 // single barrier per step
        hCur = hNext;
    };

    for (int t = 0; t < TSTEPS; t += 2) {   // TSTEPS is even
        step(t,     stA, stB);
        step(t + 1, stB, stA);
    }
}

extern "C" void kernel_launch(void* const* d_in, const int* in_sizes, int n_in,
                              void* d_out, int out_size, void* d_ws, size_t ws_size,
                              hipStream_t stream) {
    const float* x    = (const float*)d_in[0];  // [32,2048,128]
    const float* V    = (const float*)d_in[1];  // [128,256]
    const float* W    = (const float*)d_in[2];  // [256,256]
    const float* bias = (const float*)d_in[3];  // [256]
    const float* x0   = (const float*)d_in[4];  // [256]
    float* out = (float*)d_out;                 // [32,2048,256]

    (void)d_ws; (void)ws_size; (void)in_sizes; (void)n_in; (void)out_size;

    // Phase 1: 65536 tiles / 8 waves per block = 8192 blocks.
    asym_rnn_proj_wmma<<<8192, 256, 0, stream>>>(x, V, bias, out);

    // Phase 2: one WG per batch element, ~263 KB dynamic LDS.
    const size_t ldsBytes = (size_t)(UNITS * LDSTRIDE + 2 * UNITS) * sizeof(float);
    asym_rnn_scan<<<BATCH, 256, ldsBytes, stream>>>(W, x0, out);
}